// PredictHandler_87007447482376
// MI455X (gfx1250) — compile-verified
//
#include <hip/hip_runtime.h>

// ---------------------------------------------------------------------------
// RetinaFace post-process for MI455X (gfx1250, wave32).
// Pipeline: score/hist -> radix-select threshold -> gather candidates ->
// 16K bitonic sort in 128KB LDS (one WGP) -> gather/decode top-8192 ->
// tiled IoU suppression bitmask (WMMA f32 16x16x4 computes area_i+area_j
// outer sum per 16x16 tile; boxes staged via GLOBAL_LOAD_ASYNC_TO_LDS_B128)
// -> windowed greedy NMS scan (with global_prefetch) -> apply mask.
// ---------------------------------------------------------------------------

#define CONF_THR  0.5f
#define NMS_THR   0.3f
#define TOPK      8192
#define NBINS     2048
#define CANDCAP   16384

typedef __attribute__((ext_vector_type(2))) float v2f;
typedef __attribute__((ext_vector_type(8))) float v8f;

// workspace byte offsets
static constexpr size_t O_SCORES = 0;                              // 100000 f32
static constexpr size_t O_HIST   = 400000;                         // 2048 u32
static constexpr size_t O_META   = O_HIST + (size_t)NBINS * 4;     // [0]=cand cnt, [1]=thr bin
static constexpr size_t O_CAND   = O_META + 64;                    // 16384 u64
static constexpr size_t O_SELIDX = O_CAND + (size_t)CANDCAP * 8;   // 8192 u32
static constexpr size_t O_SELSC  = O_SELIDX + (size_t)TOPK * 4;    // 8192 f32
static constexpr size_t O_MASK   = ((O_SELSC + (size_t)TOPK * 4 + 255) / 256) * 256; // 8192x256 u32
static constexpr size_t O_KEEP   = O_MASK + (size_t)TOPK * 256 * 4;                  // 256 u32

// LDS byte offset of a __shared__ object: flat LDS address low 32 bits
// (ISA: the LDS aperture offset is the address's low 32 bits).
__device__ __forceinline__ unsigned lds_offset_of(const void* p) {
    return (unsigned)(unsigned long long)p;
}

// ---------------------------------------------------------------------------
__global__ void k_init(unsigned char* ws) {
    int t = blockIdx.x * blockDim.x + threadIdx.x;
    unsigned long long* cand = (unsigned long long*)(ws + O_CAND);
    unsigned* hist = (unsigned*)(ws + O_HIST);
    unsigned* meta = (unsigned*)(ws + O_META);
    unsigned* keep = (unsigned*)(ws + O_KEEP);
    if (t < CANDCAP) cand[t] = ~0ull;
    if (t < NBINS)   hist[t] = 0u;
    if (t < 4)       meta[t] = 0u;
    if (t < 256)     keep[t] = 0u;
}

// softmax(2-class) == sigmoid(c1-c0); threshold; histogram of survivors
__global__ void k_score(const float* __restrict__ conf, unsigned char* ws, int n) {
    int t = blockIdx.x * blockDim.x + threadIdx.x;
    if (t >= n) return;
    float c0 = conf[2 * t], c1 = conf[2 * t + 1];
    float sc = 1.0f / (1.0f + __expf(c0 - c1));
    float st = (sc >= CONF_THR) ? sc : 0.0f;
    ((float*)(ws + O_SCORES))[t] = st;
    if (st >= CONF_THR) {
        int b = (int)((st - CONF_THR) * 4096.0f);
        if (b > NBINS - 1) b = NBINS - 1;
        if (b < 0) b = 0;
        atomicAdd(&((unsigned*)(ws + O_HIST))[b], 1u);
    }
}

// scan histogram from top to find bin containing the TOPK-th score
__global__ void k_thresh(unsigned char* ws) {
    if (threadIdx.x != 0) return;
    unsigned* hist = (unsigned*)(ws + O_HIST);
    unsigned* meta = (unsigned*)(ws + O_META);
    unsigned cum = 0;
    int b = NBINS - 1;
    for (; b >= 0; --b) { cum += hist[b]; if (cum >= TOPK) break; }
    if (b < 0) b = 0;
    meta[1] = (unsigned)b;
}

// gather all (score,idx) with score bin >= threshold bin
__global__ void k_cand(unsigned char* ws, int n) {
    int t = blockIdx.x * blockDim.x + threadIdx.x;
    if (t >= n) return;
    float st = ((const float*)(ws + O_SCORES))[t];
    if (st < CONF_THR) return;
    int b = (int)((st - CONF_THR) * 4096.0f);
    if (b > NBINS - 1) b = NBINS - 1;
    if (b < 0) b = 0;
    unsigned* meta = (unsigned*)(ws + O_META);
    if ((unsigned)b < meta[1]) return;
    unsigned p = atomicAdd(&meta[0], 1u);
    if (p < CANDCAP) {
        unsigned ub = __float_as_uint(st);
        // key = (~scorebits, idx): ascending sort => score desc, idx asc on ties
        ((unsigned long long*)(ws + O_CAND))[p] =
            ((unsigned long long)(0xFFFFFFFFu - ub) << 32) | (unsigned)t;
    }
}

// single-workgroup bitonic sort of 16384 keys in 128KB LDS (CDNA5: 320KB/WGP)
__global__ void k_sort(unsigned char* ws) {
    extern __shared__ unsigned long long sm[];
    unsigned long long* cand = (unsigned long long*)(ws + O_CAND);
    int tid = threadIdx.x;
    for (int i = tid; i < CANDCAP; i += 1024) sm[i] = cand[i];
    __syncthreads();
    for (int k = 2; k <= CANDCAP; k <<= 1) {
        for (int j = k >> 1; j > 0; j >>= 1) {
            for (int i = tid; i < CANDCAP; i += 1024) {
                int l = i ^ j;
                if (l > i) {
                    unsigned long long a = sm[i], b = sm[l];
                    bool up = (i & k) == 0;
                    if ((a > b) == up) { sm[i] = b; sm[l] = a; }
                }
            }
            __syncthreads();
        }
    }
    unsigned* selidx = (unsigned*)(ws + O_SELIDX);
    float* selsc = (float*)(ws + O_SELSC);
    for (int i = tid; i < TOPK; i += 1024) {
        unsigned long long key = sm[i];
        unsigned ub = 0xFFFFFFFFu - (unsigned)(key >> 32);
        selsc[i] = __uint_as_float(ub);
        selidx[i] = (unsigned)key;
    }
}

// decode selected boxes/keypoints straight into d_out
__global__ void k_gather(const float* __restrict__ loc, const float* __restrict__ lm,
                         const float* __restrict__ anch, const unsigned char* ws,
                         float* __restrict__ out, int n) {
    int t = blockIdx.x * blockDim.x + threadIdx.x;
    if (t >= TOPK) return;
    unsigned a = ((const unsigned*)(ws + O_SELIDX))[t];
    float sc = ((const float*)(ws + O_SELSC))[t];
    float bx[4] = {0.f, 0.f, 0.f, 0.f};
    float kp[10] = {0.f, 0.f, 0.f, 0.f, 0.f, 0.f, 0.f, 0.f, 0.f, 0.f};
    if (a < (unsigned)n) {
        float cx = anch[a * 4 + 0], cy = anch[a * 4 + 1];
        float aw = anch[a * 4 + 2], ah = anch[a * 4 + 3];
        float lx = loc[a * 4 + 0], ly = loc[a * 4 + 1];
        float lw = loc[a * 4 + 2], lh = loc[a * 4 + 3];
        float x = cx + lx * 0.1f * aw;
        float y = cy + ly * 0.1f * ah;
        float w = aw * __expf(lw * 0.2f);
        float h = ah * __expf(lh * 0.2f);
        bx[0] = x - 0.5f * w; bx[1] = y - 0.5f * h;
        bx[2] = x + 0.5f * w; bx[3] = y + 0.5f * h;
        for (int p = 0; p < 5; ++p) {
            kp[2 * p + 0] = cx + lm[a * 10 + 2 * p + 0] * 0.1f * aw;
            kp[2 * p + 1] = cy + lm[a * 10 + 2 * p + 1] * 0.1f * ah;
        }
    } else {
        sc = 0.0f;
    }
    float* ob = out;                  // 8192 x 4
    float* ok = out + TOPK * 4;       // 8192 x 10
    float* os = out + TOPK * 14;      // 8192
    for (int c = 0; c < 4; ++c) ob[t * 4 + c] = bx[c];
    for (int c = 0; c < 10; ++c) ok[t * 10 + c] = kp[c];
    os[t] = sc;
}

// Each wave computes one 16x16 IoU tile. Boxes are staged into LDS with
// GLOBAL_LOAD_ASYNC_TO_LDS_B128 (ASYNCcnt path). The denominator outer-sum
// area_i + area_j is done with V_WMMA_F32_16X16X4_F32 (rank-2 matmul):
// A(16x4) = [area_row | 1 | 0 | 0], B(4x16) = [ones ; area_col ; 0 ; 0].
__global__ __launch_bounds__(256) void k_iou(const float* __restrict__ out,
                                             unsigned char* ws) {
    __shared__ float rbs[16 * 4];    // 16 row boxes
    __shared__ float cbs[128 * 4];   // 128 col boxes (8 tiles of 16)
    __shared__ float ra[16];
    __shared__ float ca[128];
    const int tid = threadIdx.x;
    const int wave = tid >> 5, lane = tid & 31;
    const int ti = blockIdx.x >> 6;            // 512 tile-rows
    const int tjBase = (blockIdx.x & 63) << 3; // 8 tiles of 16 cols per block

    // --- async stage: one 16B box per lane, memory -> LDS, no VGPR transit ---
    if (tid < 16) {
        unsigned ldsoff = lds_offset_of(&rbs[tid * 4]);
        unsigned goff = (unsigned)((ti * 16 + tid) * 16);
        asm volatile("global_load_async_to_lds_b128 %0, %1, %2"
                     :: "v"(ldsoff), "v"(goff), "s"(out) : "memory");
    }
    if (tid < 128) {
        unsigned ldsoff = lds_offset_of(&cbs[tid * 4]);
        unsigned goff = (unsigned)((tjBase * 16 + tid) * 16);
        asm volatile("global_load_async_to_lds_b128 %0, %1, %2"
                     :: "v"(ldsoff), "v"(goff), "s"(out) : "memory");
    }
    asm volatile("s_wait_asynccnt 0x0" ::: "memory");
    __syncthreads();

    if (tid < 16)
        ra[tid] = fmaxf(rbs[tid * 4 + 2] - rbs[tid * 4 + 0], 0.0f) *
                  fmaxf(rbs[tid * 4 + 3] - rbs[tid * 4 + 1], 0.0f);
    if (tid < 128)
        ca[tid] = fmaxf(cbs[tid * 4 + 2] - cbs[tid * 4 + 0], 0.0f) *
                  fmaxf(cbs[tid * 4 + 3] - cbs[tid * 4 + 1], 0.0f);
    __syncthreads();

    const int tj = tjBase + wave;
    // A operand (16x4 f32, 2 VGPRs): lanes 0-15 carry K=0,1; lanes 16-31 K=2,3 (zero)
    float a0 = (lane < 16) ? ra[lane & 15] : 0.0f;
    float a1 = (lane < 16) ? 1.0f : 0.0f;
    // B operand (4x16 f32, 2 VGPRs): lanes 0-15 carry K=0,1 rows; K=2,3 zero
    float b0 = (lane < 16) ? 1.0f : 0.0f;
    float b1 = (lane < 16) ? ca[wave * 16 + (lane & 15)] : 0.0f;
    v2f A = {a0, a1};
    v2f B = {b0, b1};
    v8f acc = {};
    acc = __builtin_amdgcn_wmma_f32_16x16x4_f32(
        /*neg_a=*/false, A, /*neg_b=*/false, B,
        /*c_mod=*/(short)0, acc, /*reuse_a=*/false, /*reuse_b=*/false);
    // acc[r] = area_row[r + 8*(lane>=16)] + area_col[lane&15]

    const int n = lane & 15, half = lane >> 4;
    const float cx1 = cbs[(wave * 16 + n) * 4 + 0], cy1 = cbs[(wave * 16 + n) * 4 + 1];
    const float cx2 = cbs[(wave * 16 + n) * 4 + 2], cy2 = cbs[(wave * 16 + n) * 4 + 3];
    unsigned short* m16 = (unsigned short*)(ws + O_MASK);
    #pragma unroll
    for (int r = 0; r < 8; ++r) {
        int il = r + 8 * half;
        float w = fminf(rbs[il * 4 + 2], cx2) - fmaxf(rbs[il * 4 + 0], cx1);
        w = fmaxf(w, 0.0f);
        float h = fminf(rbs[il * 4 + 3], cy2) - fmaxf(rbs[il * 4 + 1], cy1);
        h = fmaxf(h, 0.0f);
        float inter = w * h;
        float iou = inter / (acc[r] - inter + 1e-9f);
        unsigned b32 = (unsigned)__ballot(iou > NMS_THR);
        // low 16 bits: row (ti*16+r), cols of this tile; high 16: row (ti*16+r+8)
        if (lane == 0)
            m16[(size_t)(ti * 16 + r) * 512 + tj] = (unsigned short)(b32 & 0xFFFFu);
        if (lane == 16)
            m16[(size_t)(ti * 16 + r + 8) * 512 + tj] = (unsigned short)(b32 >> 16);
    }
}

// Greedy NMS scan, 32 decisions per window resolved serially by one lane,
// then parallel row-OR of kept rows into the removed bitmask. 256 windows.
// global_prefetch hides the next suppression row behind the current OR.
__global__ __launch_bounds__(256) void k_scan(const float* __restrict__ out,
                                              unsigned char* ws) {
    __shared__ unsigned removed[256];
    __shared__ unsigned kwWords[256];
    __shared__ unsigned keptList[32];
    __shared__ int keptCnt;
    const int tid = threadIdx.x;
    removed[tid] = 0u;
    kwWords[tid] = 0u;
    const unsigned* mask = (const unsigned*)(ws + O_MASK);
    const float* s = out + TOPK * 14;
    __syncthreads();
    for (int wd = 0; wd < 256; ++wd) {
        if (tid == 0) {
            unsigned r = removed[wd];
            unsigned kbits = 0;
            int cnt = 0;
            for (int b = 0; b < 32; ++b) {
                int i = wd * 32 + b;
                if (!((r >> b) & 1u) && s[i] > 0.0f) {
                    kbits |= 1u << b;
                    keptList[cnt++] = (unsigned)i;
                    r |= mask[(size_t)i * 256 + wd]; // in-window suppression
                }
            }
            kwWords[wd] = kbits;
            keptCnt = cnt;
        }
        __syncthreads();
        int cnt = keptCnt;
        for (int q = 0; q < cnt; ++q) {
            unsigned i = keptList[q];
            if (q + 1 < cnt)
                __builtin_prefetch(&mask[(size_t)keptList[q + 1] * 256 + tid], 0, 3);
            removed[tid] |= mask[(size_t)i * 256 + tid];
        }
        __syncthreads();
    }
    ((unsigned*)(ws + O_KEEP))[tid] = kwWords[tid];
}

__global__ void k_apply(const unsigned char* ws, float* __restrict__ out) {
    int t = blockIdx.x * blockDim.x + threadIdx.x;
    if (t >= TOPK) return;
    unsigned kwv = ((const unsigned*)(ws + O_KEEP))[t >> 5];
    float f = ((kwv >> (t & 31)) & 1u) ? 1.0f : 0.0f;
    for (int c = 0; c < 4; ++c) out[t * 4 + c] *= f;
    for (int c = 0; c < 10; ++c) out[TOPK * 4 + t * 10 + c] *= f;
    out[TOPK * 14 + t] *= f;
    out[TOPK * 15 + t] = f;
}

// ---------------------------------------------------------------------------
extern "C" void kernel_launch(void* const* d_in, const int* in_sizes, int n_in,
                              void* d_out, int out_size, void* d_ws, size_t ws_size,
                              hipStream_t stream) {
    const float* p_loc    = (const float*)d_in[0];
    const float* p_conf   = (const float*)d_in[1];
    const float* p_landms = (const float*)d_in[2];
    const float* anchors  = (const float*)d_in[3];
    const int n = in_sizes[1] / 2; // N_ANCH
    unsigned char* ws = (unsigned char*)d_ws;
    float* out = (float*)d_out;

    k_init  <<<(CANDCAP + 255) / 256, 256, 0, stream>>>(ws);
    k_score <<<(n + 255) / 256, 256, 0, stream>>>(p_conf, ws, n);
    k_thresh<<<1, 1, 0, stream>>>(ws);
    k_cand  <<<(n + 255) / 256, 256, 0, stream>>>(ws, n);
    k_sort  <<<1, 1024, (size_t)CANDCAP * 8, stream>>>(ws);
    k_gather<<<(TOPK + 255) / 256, 256, 0, stream>>>(p_loc, p_landms, anchors, ws, out, n);
    k_iou   <<<(512 * 512) / 8, 256, 0, stream>>>(out, ws);
    k_scan  <<<1, 256, 0, stream>>>(out, ws);
    k_apply <<<(TOPK + 255) / 256, 256, 0, stream>>>(ws, out);
}